// NeuralMeshRenderer_13426067768025
// MI455X (gfx1250) — compile-verified
//
#include <hip/hip_runtime.h>

#define IMG 256
#define ORIGS 1024.0f
#define NEARP 0.1f
#define FARP 100.0f
#define BEPS 1e-8f

typedef __attribute__((ext_vector_type(2))) float v2f;
typedef __attribute__((ext_vector_type(8))) float v8f;

// fast reciprocal: v_rcp_f32 (~1 ulp) + one Newton-Raphson step (~0.5 ulp)
// replaces the ~10-op IEEE div sequence in the per-pixel hot loop
__device__ __forceinline__ float fast_rcp(float x) {
    float r = __builtin_amdgcn_rcpf(x);
    r = fmaf(fmaf(-x, r, 1.0f), r, r);
    return r;
}

// ---------------------------------------------------------------- init z-buffer
__global__ void nmr_init_zbuf(unsigned int* __restrict__ z, int n) {
    int i = blockIdx.x * blockDim.x + threadIdx.x;
    if (i < n) z[i] = 0x42C80000u; // bits of 100.0f (FAR)
}

// ---------------------------------------------------------------- projection
__global__ void nmr_project(const float* __restrict__ verts,
                            const float* __restrict__ Kc,
                            const float* __restrict__ Rc,
                            const float* __restrict__ tc,
                            const float* __restrict__ dc,
                            float4* __restrict__ ndc, int B, int V) {
    int i = blockIdx.x * blockDim.x + threadIdx.x;
    if (i >= B * V) return;
    int b = i / V;
    const float* p  = verts + (size_t)i * 3;
    const float* Rb = Rc + b * 9;
    const float* Kb = Kc + b * 9;
    const float* tb = tc + b * 3;
    const float* db = dc + b * 5;
    float X = p[0], Y = p[1], Z = p[2];
    float x = Rb[0]*X + Rb[1]*Y + Rb[2]*Z + tb[0];
    float y = Rb[3]*X + Rb[4]*Y + Rb[5]*Z + tb[1];
    float z = Rb[6]*X + Rb[7]*Y + Rb[8]*Z + tb[2];
    const float eps = 1e-9f;
    float rz = 1.0f / (z + eps);
    float x_ = x * rz, y_ = y * rz;
    float k1 = db[0], k2 = db[1], p1 = db[2], p2 = db[3], k3 = db[4];
    float r2 = x_ * x_ + y_ * y_;
    float rad = 1.0f + k1 * r2 + k2 * r2 * r2 + k3 * r2 * r2 * r2;
    float x__ = x_ * rad + 2.0f * p1 * x_ * y_ + p2 * (r2 + 2.0f * x_ * x_);
    float y__ = y_ * rad + p1 * (r2 + 2.0f * y_ * y_) + 2.0f * p2 * x_ * y_;
    float u  = Kb[0] * x__ + Kb[1] * y__ + Kb[2];
    float vc = Kb[3] * x__ + Kb[4] * y__ + Kb[5];
    vc = ORIGS - vc;
    u  = 2.0f * (u  - ORIGS * 0.5f) / ORIGS;
    vc = 2.0f * (vc - ORIGS * 0.5f) / ORIGS;
    ndc[i] = make_float4(u, vc, z, 0.0f);
}

// ---------------------------------------------------------------- rasterizer
// One wave32 per face. Per 16x16 pixel tile, evaluate the three affine planes
// (w0, w1, zinv) with V_WMMA_F32_16X16X4_F32:
//   D[m][n] = A[m][0]*B[0][n] + C[m][n] = 1*(a*x_n + c) + b*y_m
// then each lane tests its 8 pixels and atomic-mins the z-buffer.
__global__ void nmr_raster(const int* __restrict__ faces,
                           const float4* __restrict__ ndc,
                           unsigned int* __restrict__ zbuf,
                           int B, int V, int F) {
    int wid  = (blockIdx.x * blockDim.x + threadIdx.x) >> 5;
    int lane = threadIdx.x & 31;
    if (wid >= B * F) return;                 // wave-uniform
    int b = wid / F, f = wid - b * F;

    const int* fp = faces + ((size_t)(b * F + f)) * 3;
    int i0 = fp[0], i1 = fp[1], i2 = fp[2];
    const float4* vb = ndc + (size_t)b * V;
    float4 q0 = vb[i0], q1 = vb[i1], q2 = vb[i2];

    // reference: any z <= eps  -> face contributes only FAR -> no-op
    if (!(q0.z > BEPS && q1.z > BEPS && q2.z > BEPS)) return;
    float denom = (q1.y - q2.y) * (q0.x - q2.x) + (q2.x - q1.x) * (q0.y - q2.y);
    if (!(fabsf(denom) > BEPS)) return;       // degenerate -> no-op

    float rd = 1.0f / denom;                  // per-face: keep exact division
    float a0 = (q1.y - q2.y) * rd;
    float b0 = (q2.x - q1.x) * rd;
    float c0 = -(a0 * q2.x + b0 * q2.y);
    float a1 = (q2.y - q0.y) * rd;
    float b1 = (q0.x - q2.x) * rd;
    float c1 = -(a1 * q2.x + b1 * q2.y);
    float rz0 = 1.0f / q0.z, rz1 = 1.0f / q1.z, rz2 = 1.0f / q2.z;
    float d0 = rz0 - rz2, d1 = rz1 - rz2;
    float az = a0 * d0 + a1 * d1;
    float bz = b0 * d0 + b1 * d1;
    float cz = c0 * d0 + c1 * d1 + rz2;

    // bounding box in pixel indices (pixel i center = (2i+1-IMG)/IMG)
    float xmn = fminf(q0.x, fminf(q1.x, q2.x));
    float xmx = fmaxf(q0.x, fmaxf(q1.x, q2.x));
    float ymn = fminf(q0.y, fminf(q1.y, q2.y));
    float ymx = fmaxf(q0.y, fmaxf(q1.y, q2.y));
    int ixlo = max(0,       (int)ceilf (128.0f * xmn + 127.5f - 1e-3f));
    int ixhi = min(IMG - 1, (int)floorf(128.0f * xmx + 127.5f + 1e-3f));
    int iylo = max(0,       (int)ceilf (128.0f * ymn + 127.5f - 1e-3f));
    int iyhi = min(IMG - 1, (int)floorf(128.0f * ymx + 127.5f + 1e-3f));
    if (ixlo > ixhi || iylo > iyhi) return;   // off-screen (wave-uniform)

    int  laneM = lane & 15;
    int  hi    = lane >> 4;                   // lanes 16-31 -> rows 8-15, A rows K=2,3 (zero)
    float Ax   = hi ? 0.0f : 1.0f;
    unsigned int* zb = zbuf + ((size_t)b << 16);
    const float step = 2.0f / (float)IMG;

    v2f Av; Av.x = Ax; Av.y = 0.0f;

    for (int ty = iylo >> 4; ty <= iyhi >> 4; ++ty) {
        float py0 = ((float)((ty << 4) + (hi << 3)) * 2.0f + 1.0f - (float)IMG) * (1.0f / IMG);
        for (int tx = ixlo >> 4; tx <= ixhi >> 4; ++tx) {
            float px = ((float)((tx << 4) + laneM) * 2.0f + 1.0f - (float)IMG) * (1.0f / IMG);

            v8f c0v, c1v, czv;
#pragma unroll
            for (int j = 0; j < 8; ++j) {
                float py = py0 + step * (float)j;
                c0v[j] = b0 * py;
                c1v[j] = b1 * py;
                czv[j] = bz * py;
            }
            v2f B0; B0.x = hi ? 0.0f : fmaf(a0, px, c0); B0.y = 0.0f;
            v2f B1; B1.x = hi ? 0.0f : fmaf(a1, px, c1); B1.y = 0.0f;
            v2f Bz; Bz.x = hi ? 0.0f : fmaf(az, px, cz); Bz.y = 0.0f;

            v8f w0v = __builtin_amdgcn_wmma_f32_16x16x4_f32(false, Av, false, B0,
                                                            (short)0, c0v, false, false);
            v8f w1v = __builtin_amdgcn_wmma_f32_16x16x4_f32(false, Av, false, B1,
                                                            (short)0, c1v, false, false);
            v8f zvv = __builtin_amdgcn_wmma_f32_16x16x4_f32(false, Av, false, Bz,
                                                            (short)0, czv, false, false);

            int col  = (tx << 4) + laneM;
            int rowb = (ty << 4) + (hi << 3);
#pragma unroll
            for (int j = 0; j < 8; ++j) {
                float w0 = w0v[j], w1 = w1v[j];
                float w2 = 1.0f - w0 - w1;
                float zi = zvv[j];
                float zp = (zi > BEPS) ? fast_rcp(zi) : 1.0f;   // cheap rcp+NR
                bool valid = (w0 >= 0.0f) && (w1 >= 0.0f) && (w2 >= 0.0f) &&
                             (zp > NEARP) && (zp < FARP);
                if (valid)
                    atomicMin(&zb[((rowb + j) << 8) + col], __float_as_uint(zp));
            }
        }
    }
}

// ---------------------------------------------------------------- vertical flip (in place)
__global__ void nmr_flip(float* __restrict__ out, int B) {
    int i = blockIdx.x * blockDim.x + threadIdx.x;
    int total = B * (IMG / 2) * IMG;
    if (i >= total) return;
    int b = i / ((IMG / 2) * IMG);
    int rem = i - b * (IMG / 2) * IMG;
    int y = rem >> 8;
    int x = rem & (IMG - 1);
    size_t idx1 = ((size_t)b << 16) | ((size_t)y << 8) | (size_t)x;
    size_t idx2 = ((size_t)b << 16) | ((size_t)(IMG - 1 - y) << 8) | (size_t)x;
    float t1 = out[idx1];
    float t2 = out[idx2];
    out[idx1] = t2;
    out[idx2] = t1;
}

// ---------------------------------------------------------------- launcher
extern "C" void kernel_launch(void* const* d_in, const int* in_sizes, int n_in,
                              void* d_out, int out_size, void* d_ws, size_t ws_size,
                              hipStream_t stream) {
    const float* verts = (const float*)d_in[0];   // (B,V,3)
    const int*   faces = (const int*)  d_in[1];   // (B,F,3)
    const float* Kc    = (const float*)d_in[2];   // (B,3,3)
    const float* Rc    = (const float*)d_in[3];   // (B,3,3)
    const float* tc    = (const float*)d_in[4];   // (B,1,3)
    const float* dc    = (const float*)d_in[5];   // (B,5)

    int B  = in_sizes[5] / 5;
    int BV = in_sizes[0] / 3;
    int BF = in_sizes[1] / 3;
    int V  = BV / B;
    int F  = BF / B;   // doubled faces are winding-mirrors -> identical zp, min idempotent

    // d_out doubles as the uint-bits z-buffer (positive-float order == uint order)
    unsigned int* zbuf = (unsigned int*)d_out;
    float4* ndc = (float4*)d_ws;                 // B*V float4 = 320 KB

    int nz = B * IMG * IMG;
    nmr_init_zbuf<<<(nz + 255) / 256, 256, 0, stream>>>(zbuf, nz);
    nmr_project<<<(BV + 255) / 256, 256, 0, stream>>>(verts, Kc, Rc, tc, dc, ndc, B, V);

    int threads = BF * 32;                       // one wave32 per face
    nmr_raster<<<(threads + 255) / 256, 256, 0, stream>>>(faces, ndc, zbuf, B, V, F);

    int nflip = B * (IMG / 2) * IMG;
    nmr_flip<<<(nflip + 255) / 256, 256, 0, stream>>>((float*)d_out, B);
}